// Path_76828374991393
// MI455X (gfx1250) — compile-verified
//
#include <hip/hip_runtime.h>
#include <math.h>

#ifndef TPB
#define TPB 256
#endif
#define EPS_F 1e-4f
#define INV_PI_F 0.31830988618379067f

// CDNA5 async global->LDS path (fire-and-forget staging, tracked by ASYNCcnt).
#if defined(__HIP_DEVICE_COMPILE__) &&                                   \
    __has_builtin(__builtin_amdgcn_global_load_async_to_lds_b64) &&      \
    __has_builtin(__builtin_amdgcn_s_wait_asynccnt)
#define USE_ASYNC_LDS 1
#else
#define USE_ASYNC_LDS 0
#endif

#if USE_ASYNC_LDS
// Builtin signature (from hipcc diagnostic): pointee is a 2 x int GCC vector.
typedef int v2i_t __attribute__((vector_size(2 * sizeof(int))));
typedef __attribute__((address_space(1))) v2i_t* as1_v2i_t;
typedef __attribute__((address_space(3))) v2i_t* as3_v2i_t;
#endif

__device__ __forceinline__ float rcp_f (float x) { return __builtin_amdgcn_rcpf(x);  }
__device__ __forceinline__ float rsq_f (float x) { return __builtin_amdgcn_rsqf(x);  }
__device__ __forceinline__ float sqrt_f(float x) { return __builtin_amdgcn_sqrtf(x); }
// v_sin_f32/v_cos_f32 compute sin/cos(2*pi*x): feed u2 directly (phi = 2*pi*u2).
__device__ __forceinline__ float sin2pi(float x) { return __builtin_amdgcn_sinf(x); }
__device__ __forceinline__ float cos2pi(float x) { return __builtin_amdgcn_cosf(x); }

// Ray-unit-sphere intersection, faithful to the JAX reference.
__device__ __forceinline__ bool sphere_isect(
    float ox, float oy, float oz, float dx, float dy, float dz,
    float& px, float& py, float& pz, float& nx, float& ny, float& nz)
{
    float b    = ox * dx + oy * dy + oz * dz;
    float c    = ox * ox + oy * oy + oz * oz - 1.0f;
    float disc = b * b - c;
    float sq   = sqrt_f(fmaxf(disc, 0.0f));
    float t0   = -b - sq;
    float t1   = -b + sq;
    float t    = (t0 > EPS_F) ? t0 : t1;
    bool  hit  = (disc > 0.0f) && (t > EPS_F);
    px = fmaf(t, dx, ox);
    py = fmaf(t, dy, oy);
    pz = fmaf(t, dz, oz);
    float pl  = sqrt_f(px * px + py * py + pz * pz);
    float inv = rcp_f(fmaxf(pl, 1e-12f));
    nx = px * inv; ny = py * inv; nz = pz * inv;
    return hit;
}

__global__ __launch_bounds__(TPB) void path_trace_kernel(
    const float* __restrict__ ray_o,
    const float* __restrict__ ray_d,
    const float* __restrict__ u_bsdf,   // (4, N, 2) f32
    const float* __restrict__ albedo,   // (3,)
    const float* __restrict__ light,    // (3,)
    float* __restrict__ out_res,        // (N, 3)
    float* __restrict__ out_act,        // (N,) as 0.0/1.0
    int N)
{
#if USE_ASYNC_LDS
    __shared__ float2 ldsU[3 * TPB];    // 3 bounce planes * 256 threads * 8B = 6KB
#endif
    const int tid = threadIdx.x;
    const int i   = blockIdx.x * TPB + tid;
    if (i >= N) return;

    const float2* up = (const float2*)u_bsdf;   // plane p, ray i -> up[p*N + i]

#if USE_ASYNC_LDS
    // Stage the 3 live u_bsdf planes into LDS asynchronously; the transfers
    // overlap the camera-ray setup + first intersection below. (The depth-3
    // samples never influence the output, so plane 3 is not fetched.)
    #pragma unroll
    for (int pl = 0; pl < 3; ++pl) {
        __builtin_amdgcn_global_load_async_to_lds_b64(
            (as1_v2i_t)(up + (size_t)pl * N + i),
            (as3_v2i_t)(&ldsU[pl * TPB + tid]),
            /*offset=*/0, /*cpol=*/0);
    }
#else
    __builtin_prefetch(up + (size_t)0 * N + i, 0, 3);   // global_prefetch_b8
    __builtin_prefetch(up + (size_t)1 * N + i, 0, 3);
    __builtin_prefetch(up + (size_t)2 * N + i, 0, 3);
#endif

    // Uniform scene constants.
    const float alx = albedo[0], aly = albedo[1], alz = albedo[2];
    const float lix = light[0],  liy = light[1],  liz = light[2];
    const float cax = fmaxf(alx, 1e-10f);   // clip(weight, 1e-10, None)
    const float cay = fmaxf(aly, 1e-10f);
    const float caz = fmaxf(alz, 1e-10f);

    // Camera ray: o = ray_o*0.5 + (0,0,-4); d = normalize(ray_d*0.2 + (0,0,1))
    float ox = ray_o[3 * i + 0] * 0.5f;
    float oy = ray_o[3 * i + 1] * 0.5f;
    float oz = ray_o[3 * i + 2] * 0.5f - 4.0f;
    float dx = ray_d[3 * i + 0] * 0.2f;
    float dy = ray_d[3 * i + 1] * 0.2f;
    float dz = ray_d[3 * i + 2] * 0.2f + 1.0f;
    {
        float dl  = sqrt_f(dx * dx + dy * dy + dz * dz);
        float inv = rcp_f(fmaxf(dl, 1e-12f));
        dx *= inv; dy *= inv; dz *= inv;
    }

    float px, py, pz, nx, ny, nz;
    bool active = sphere_isect(ox, oy, oz, dx, dy, dz, px, py, pz, nx, ny, nz);
    const bool original_active = active;

    float tpx = 1.0f, tpy = 1.0f, tpz = 1.0f;   // throughput
    float rx = 0.0f, ry = 0.0f, rz = 0.0f;      // accumulated result

#if USE_ASYNC_LDS
    __builtin_amdgcn_s_wait_asynccnt(0);        // staged planes now resident
#endif

    #pragma unroll
    for (int depth = 0; depth < 4; ++depth) {
        // ---- direct lighting from point light at (3,4,5) ----
        float lxv = 3.0f - px, lyv = 4.0f - py, lzv = 5.0f - pz;
        float dist2 = lxv * lxv + lyv * lyv + lzv * lzv;
        float invd  = rsq_f(dist2);                       // 1/sqrt(dist2)
        float cosl  = fmaxf((lxv * nx + lyv * ny + lzv * nz) * invd, 0.0f);
        float k     = cosl * INV_PI_F * (invd * invd);    // cos_l/(pi*dist2)
        float gate  = active ? k : 0.0f;
        rx = fmaf(tpx * alx * lix, gate, rx);
        ry = fmaf(tpy * aly * liy, gate, ry);
        rz = fmaf(tpz * alz * liz, gate, rz);

        if (depth == 3) break;   // depth-3 bounce never affects the output

        // ---- shading frame (branchless Duff et al., as in reference) ----
        float s   = (nz >= 0.0f) ? 1.0f : -1.0f;
        float a   = -rcp_f(s + nz);          // |s+nz| >= 1, safe
        float bb  = nx * ny * a;
        float t1x = fmaf(s * nx * nx, a, 1.0f), t1y = s * bb,              t1z = -s * nx;
        float t2x = bb,                          t2y = fmaf(ny * ny, a, s), t2z = -ny;

        // ---- cosine-weighted hemisphere sample ----
#if USE_ASYNC_LDS
        float2 uu = ldsU[depth * TPB + tid];
#else
        float2 uu = up[(size_t)depth * N + i];
#endif
        float u1 = uu.x, u2 = uu.y;
        float r   = sqrt_f(u1);
        float wox = r * cos2pi(u2);                   // r*cos(2*pi*u2)
        float woy = r * sin2pi(u2);                   // r*sin(2*pi*u2)
        float woz = sqrt_f(fmaxf(1.0f - u1, 0.0f));

        // throughput *= clip(albedo, 1e-10, None)
        tpx *= cax; tpy *= cay; tpz *= caz;
        active = active && ((tpx > 0.0f) || (tpy > 0.0f) || (tpz > 0.0f));

        // new ray
        dx = wox * t1x + woy * t2x + woz * nx;
        dy = wox * t1y + woy * t2y + woz * ny;
        dz = wox * t1z + woy * t2z + woz * nz;
        ox = fmaf(EPS_F, nx, px);
        oy = fmaf(EPS_F, ny, py);
        oz = fmaf(EPS_F, nz, pz);

        float p2x, p2y, p2z, n2x, n2y, n2z;
        bool hit2 = sphere_isect(ox, oy, oz, dx, dy, dz,
                                 p2x, p2y, p2z, n2x, n2y, n2z);
        active = active && hit2;
        // p/n frozen for dead rays (matches jnp.where latching)
        px = active ? p2x : px; py = active ? p2y : py; pz = active ? p2z : pz;
        nx = active ? n2x : nx; ny = active ? n2y : ny; nz = active ? n2z : nz;
    }

    out_res[3 * i + 0] = rx;
    out_res[3 * i + 1] = ry;
    out_res[3 * i + 2] = rz;
    out_act[i] = original_active ? 1.0f : 0.0f;
}

extern "C" void kernel_launch(void* const* d_in, const int* in_sizes, int n_in,
                              void* d_out, int out_size, void* d_ws, size_t ws_size,
                              hipStream_t stream) {
    const float* ray_o  = (const float*)d_in[0];
    const float* ray_d  = (const float*)d_in[1];
    const float* u_bsdf = (const float*)d_in[2];
    const float* albedo = (const float*)d_in[3];
    const float* light  = (const float*)d_in[4];

    const int N = in_sizes[0] / 3;
    float* out_res = (float*)d_out;
    float* out_act = (float*)d_out + (size_t)3 * N;

    dim3 grid((N + TPB - 1) / TPB);
    path_trace_kernel<<<grid, TPB, 0, stream>>>(
        ray_o, ray_d, u_bsdf, albedo, light, out_res, out_act, N);
}